// RegionClassifier0_22411139350994
// MI455X (gfx1250) — compile-verified
//
#include <hip/hip_runtime.h>

#define B_    512
#define P_    65536
#define NOUT_ 512
#define COUT_ 4
#define NCLS_ 10
#define NPAD  16
#define ROWS  8
#define CHUNKS 8

typedef __attribute__((ext_vector_type(2))) float v2f;
typedef __attribute__((ext_vector_type(8))) float v8f;

// ---------------- zero h ----------------
__global__ void rc_zero(float* __restrict__ h) {
    h[blockIdx.x * 256 + threadIdx.x] = 0.0f;
}

// ---------------- Weff[j,k] = sum_c W_fgl[c,j] * W_fc[c*512+j, k] ----------------
__global__ void rc_prep_weff(const float* __restrict__ W_fgl,
                             const float* __restrict__ W_fc,
                             float* __restrict__ Weff) {
    int idx = blockIdx.x * 256 + threadIdx.x;   // 0 .. 512*16-1
    int j = idx >> 4;
    int k = idx & 15;
    float acc = 0.0f;
    if (k < NCLS_) {
#pragma unroll
        for (int c = 0; c < COUT_; ++c)
            acc += W_fgl[c * NOUT_ + j] * W_fc[(c * NOUT_ + j) * NCLS_ + k];
    }
    Weff[idx] = acc;
}

// ---------------- bias_eff[k] = b_fc[k] + sum_r b_fgl[r] * W_fc[r,k] ----------------
__global__ void rc_prep_bias(const float* __restrict__ b_fgl,
                             const float* __restrict__ W_fc,
                             const float* __restrict__ b_fc,
                             float* __restrict__ bias) {
    __shared__ float s[256][NCLS_];
    int t = threadIdx.x;
    float acc[NCLS_];
#pragma unroll
    for (int k = 0; k < NCLS_; ++k) acc[k] = 0.0f;
    for (int r = t; r < COUT_ * NOUT_; r += 256) {
        float bv = b_fgl[r];
#pragma unroll
        for (int k = 0; k < NCLS_; ++k) acc[k] += bv * W_fc[r * NCLS_ + k];
    }
#pragma unroll
    for (int k = 0; k < NCLS_; ++k) s[t][k] = acc[k];
    __syncthreads();
    for (int stride = 128; stride > 0; stride >>= 1) {
        if (t < stride) {
#pragma unroll
            for (int k = 0; k < NCLS_; ++k) s[t][k] += s[t + stride][k];
        }
        __syncthreads();
    }
    if (t < NPAD) bias[t] = (t < NCLS_) ? (s[0][t] + b_fc[t]) : 0.0f;
}

// ---------------- segment sum: h[b,j] += x[b,p] for seg[p]==j ----------------
// grid (CHUNKS, B_/ROWS), block 256. LDS-privatized bins, ds_add_f32 atomics,
// one global_atomic_add_f32 flush per bin.
__global__ void rc_segsum(const float* __restrict__ x,
                          const int* __restrict__ seg,
                          float* __restrict__ h) {
    __shared__ float bins[ROWS * NOUT_];
    int t = threadIdx.x;
    for (int i = t; i < ROWS * NOUT_; i += 256) bins[i] = 0.0f;
    __syncthreads();

    int b0 = blockIdx.y * ROWS;
    int pbase = blockIdx.x * (P_ / CHUNKS);
    const int iters = (P_ / CHUNKS) / (256 * 4);   // 8

    for (int it = 0; it < iters; ++it) {
        int p = pbase + (it * 256 + t) * 4;
        int4 s4 = *(const int4*)(seg + p);
#pragma unroll
        for (int r = 0; r < ROWS; ++r) {
            float4 x4 = *(const float4*)(x + (size_t)(b0 + r) * P_ + p);
            float* rb = bins + r * NOUT_;
            atomicAdd(rb + s4.x, x4.x);
            atomicAdd(rb + s4.y, x4.y);
            atomicAdd(rb + s4.z, x4.z);
            atomicAdd(rb + s4.w, x4.w);
        }
    }
    __syncthreads();
    for (int i = t; i < ROWS * NOUT_; i += 256) {
        int r = i >> 9;          // /512
        int j = i & (NOUT_ - 1);
        atomicAdd(&h[(size_t)(b0 + r) * NOUT_ + j], bins[i]);
    }
}

// ---------------- out = h @ Weff + bias, exact fp32 via V_WMMA_F32_16X16X4_F32 ----
// One wave (32 lanes) per 16-row batch tile. A: 16x4 fp32 (2 VGPRs, K = v + 2*grp),
// B: 4x16 fp32 (2 VGPRs, same K striping, N = lane%16), D: 16x16 (M = v + 8*grp).
__global__ void rc_gemm_wmma(const float* __restrict__ h,
                             const float* __restrict__ Weff,
                             const float* __restrict__ bias,
                             float* __restrict__ out) {
    int lane = threadIdx.x;      // 0..31
    int col  = lane & 15;        // N (class) and also A's M row selector
    int grp  = lane >> 4;
    int b0   = blockIdx.x * 16;

    const float* arow = h + (size_t)(b0 + col) * NOUT_;
    v8f c = {};
    for (int k0 = 0; k0 < NOUT_; k0 += 4) {
        int ka = k0 + 2 * grp;
        v2f a, b;
        a.x = arow[ka + 0];
        a.y = arow[ka + 1];
        b.x = Weff[(ka + 0) * NPAD + col];
        b.y = Weff[(ka + 1) * NPAD + col];
        c = __builtin_amdgcn_wmma_f32_16x16x4_f32(
                /*neg_a=*/false, a, /*neg_b=*/false, b,
                /*c_mod=*/(short)0, c, /*reuse_a=*/false, /*reuse_b=*/false);
    }

    float bk = bias[col];
    if (col < NCLS_) {
#pragma unroll
        for (int v = 0; v < 8; ++v) {
            int m = v + 8 * grp;
            out[(size_t)(b0 + m) * NCLS_ + col] = c[v] + bk;
        }
    }
}

extern "C" void kernel_launch(void* const* d_in, const int* in_sizes, int n_in,
                              void* d_out, int out_size, void* d_ws, size_t ws_size,
                              hipStream_t stream) {
    const float* x     = (const float*)d_in[0];
    const float* W_fgl = (const float*)d_in[1];
    const float* b_fgl = (const float*)d_in[2];
    const float* W_fc  = (const float*)d_in[3];
    const float* b_fc  = (const float*)d_in[4];
    const int*   seg   = (const int*)d_in[5];
    float* out = (float*)d_out;

    char* ws = (char*)d_ws;
    float* h    = (float*)ws;                                         // 512*512*4 = 1 MB
    float* Weff = (float*)(ws + (size_t)B_ * NOUT_ * 4);              // 512*16*4 = 32 KB
    float* bias = (float*)(ws + (size_t)B_ * NOUT_ * 4 + (size_t)NOUT_ * NPAD * 4);

    rc_zero<<<(B_ * NOUT_) / 256, 256, 0, stream>>>(h);
    rc_prep_weff<<<(NOUT_ * NPAD) / 256, 256, 0, stream>>>(W_fgl, W_fc, Weff);
    rc_prep_bias<<<1, 256, 0, stream>>>(b_fgl, W_fc, b_fc, bias);
    rc_segsum<<<dim3(CHUNKS, B_ / ROWS), 256, 0, stream>>>(x, seg, h);
    rc_gemm_wmma<<<B_ / 16, 32, 0, stream>>>(h, Weff, bias, out);
}